// HexCirclePool_86062554677552
// MI455X (gfx1250) — compile-verified
//
#include <hip/hip_runtime.h>
#include <hip/hip_bf16.h>

// Problem constants (from the reference)
#define B_ 64
#define C_ 128
#define N_ 8192
#define P_ 2048
#define K_ 7

// Tuning
#define THREADS 256
#define PJ      8                     // P_/THREADS p-slots per thread
#define GRID    1024
#define ITERS   4                     // (B_*C_) / (2*GRID) row-pairs per WG
#define PAIR_ELEMS (2 * N_)           // 2 rows = 16384 floats = 64 KB (contiguous)
#define LDS_BYTES  (2 * PAIR_ELEMS * 4)  // double buffer: 128 KB

typedef unsigned int v4u __attribute__((ext_vector_type(4)));
typedef int          v8i __attribute__((ext_vector_type(8)));
typedef int          v4i __attribute__((ext_vector_type(4)));

// Issue one TDM DMA: 64 KB contiguous (two x rows) global -> LDS.
// D# per CDNA5 ISA ch.8: group0 = control/addresses, group1 = dims/strides.
// clang-23 toolchain: 6-arg builtin (g0, g1, g2, g3, extra, cpol).
__device__ __forceinline__ void tdm_load_pair(const float* g, unsigned lds_off_bytes)
{
    unsigned long long ga = (unsigned long long)(const void*)g;
    v4u g0 = { 1u,                                        // count=1 (valid), user mode
               lds_off_bytes,                             // lds_addr (bytes)
               (unsigned)(ga & 0xFFFFFFFFull),            // global_addr[31:0]
               (unsigned)((ga >> 32) & 0x1FFFFFFull)      // global_addr[56:32]
                 | (2u << 30) };                          // type = 2 ("image")
    v8i g1 = { 0x00020000,                 // data_size = 2 -> 4-byte elements
               (int)(PAIR_ELEMS << 16),    // tensor_dim0[15:0] in bits[31:16]
               (int)(1 << 16),             // tensor_dim0[31:16]=0 | tensor_dim1=1
               (int)(PAIR_ELEMS << 16),    // tensor_dim1 hi=0 | tile_dim0=16384
               1,                          // tile_dim1=1, tile_dim2=0 (1-D tile)
               PAIR_ELEMS,                 // tensor_dim0_stride[31:0]
               0, 0 };                     // stride hi / tensor_dim1_stride = 0
    v4i z4 = { 0, 0, 0, 0 };               // groups 2/3 unused (<=2-D tensor)
    v8i z8 = { 0, 0, 0, 0, 0, 0, 0, 0 };   // unused trailing group
    __builtin_amdgcn_tensor_load_to_lds(g0, g1, z4, z4, z8, /*cpol=*/0);
}

__global__ __launch_bounds__(THREADS)
void HexCirclePool_gather_tdm_kernel(const float* __restrict__ x,
                                     const int* __restrict__ cluster_idx,
                                     const unsigned char* __restrict__ cluster_mask,
                                     float* __restrict__ out)
{
    extern __shared__ float lds[];     // 2 * 16384 floats (double buffer)
    const int tid = (int)threadIdx.x;
    const int wg  = (int)blockIdx.x;

    // ---- One-time: pull this thread's 8 clusters (idx, weight) into registers.
    // weight = mask ? 1/len : 0  -> folds mask AND the divide out of the hot loop.
    int   ridx[PJ][K_];
    float rw  [PJ][K_];
#pragma unroll
    for (int j = 0; j < PJ; ++j) {
        const int p = tid + THREADS * j;
        int cnt = 0;
#pragma unroll
        for (int k = 0; k < K_; ++k) cnt += (cluster_mask[p * K_ + k] != 0) ? 1 : 0;
        const float inv = 1.0f / (float)(cnt > 0 ? cnt : 1);
#pragma unroll
        for (int k = 0; k < K_; ++k) {
            ridx[j][k] = cluster_idx[p * K_ + k];
            rw[j][k]   = (cluster_mask[p * K_ + k] != 0) ? inv : 0.0f;
        }
    }

    const bool issuer = (tid < 32);    // wave 0 drives the TDM (EXEC ignored by TDM)
    if (issuer)
        tdm_load_pair(x + (size_t)(wg * ITERS + 0) * PAIR_ELEMS, 0u);

    for (int it = 0; it < ITERS; ++it) {
        if (issuer) {
            if (it + 1 < ITERS) {
                // prefetch next row-pair into the other LDS buffer, then make
                // sure the *current* pair (older DMA) has landed (cnt <= 1)
                tdm_load_pair(x + (size_t)(wg * ITERS + it + 1) * PAIR_ELEMS,
                              (unsigned)(((it + 1) & 1) * PAIR_ELEMS * 4));
                __builtin_amdgcn_s_wait_tensorcnt(1);
            } else {
                __builtin_amdgcn_s_wait_tensorcnt(0);
            }
        }
        __syncthreads();               // tile `it` now visible to all waves

        const float* lx = lds + (it & 1) * PAIR_ELEMS;   // row0 base; row1 = +N_
        float acc0[PJ], acc1[PJ];
#pragma unroll
        for (int j = 0; j < PJ; ++j) { acc0[j] = 0.0f; acc1[j] = 0.0f; }

#pragma unroll
        for (int j = 0; j < PJ; ++j) {
#pragma unroll
            for (int k = 0; k < K_; ++k) {
                const int n = ridx[j][k];
                const float wgt = rw[j][k];
                acc0[j] += lx[n]      * wgt;   // ds_load_b32 gather, row 0
                acc1[j] += lx[n + N_] * wgt;   // ds_load_b32 gather, row 1
            }
        }

        float* o = out + (size_t)(2 * (wg * ITERS + it)) * P_;
#pragma unroll
        for (int j = 0; j < PJ; ++j) {
            const int p = tid + THREADS * j;               // coalesced, streaming
            __builtin_nontemporal_store(acc0[j], o + p);
            __builtin_nontemporal_store(acc1[j], o + P_ + p);
        }
        __syncthreads();               // buffer (it&1) free for the it+2 DMA
    }
}

extern "C" void kernel_launch(void* const* d_in, const int* in_sizes, int n_in,
                              void* d_out, int out_size, void* d_ws, size_t ws_size,
                              hipStream_t stream)
{
    (void)in_sizes; (void)n_in; (void)out_size; (void)d_ws; (void)ws_size;
    const float*         x   = (const float*)d_in[0];
    const int*           ci  = (const int*)d_in[1];
    const unsigned char* cm  = (const unsigned char*)d_in[2];  // jax bool -> 1 byte
    float*               out = (float*)d_out;

    HexCirclePool_gather_tdm_kernel<<<GRID, THREADS, LDS_BYTES, stream>>>(x, ci, cm, out);
}